// AMGCN_54623394070845
// MI455X (gfx1250) — compile-verified
//
#include <hip/hip_runtime.h>
#include <hip/hip_bf16.h>

typedef __attribute__((ext_vector_type(2))) float v2f;
typedef __attribute__((ext_vector_type(8))) float v8f;

#define N_NODES 100000
#define N_GRAPHS 512
#define D_IN 300
#define D_HID 300
#define D_OUT 128
#define KMAX 300   // all three GEMMs have K == 300

// ---------------------------------------------------------------------------
// Pack W[K, Nout] (row-major) into WMMA B-fragment order with zero padding:
//   slot = ((ct*(K/4) + k4)*2 + half)*16 + l16   (each slot = float2)
//   value = ( W[k4*4 + 2*half    , ct*16 + l16],
//             W[k4*4 + 2*half + 1, ct*16 + l16] )      (0 if col >= Nout)
// ---------------------------------------------------------------------------
__global__ void gcn_pack_w(const float* __restrict__ W, float* __restrict__ Wp,
                           int K, int Nout, int nColTiles) {
    const int idx = blockIdx.x * blockDim.x + threadIdx.x;   // float2 slot id
    const int total = nColTiles * (K / 4) * 32;
    if (idx >= total) return;
    const int l16 = idx & 15;
    int t = idx >> 4;
    const int half = t & 1;  t >>= 1;
    const int k4 = t % (K / 4);
    const int ct = t / (K / 4);
    const int ks = k4 * 4 + 2 * half;
    const int c  = ct * 16 + l16;
    v2f v;
    v.x = (c < Nout) ? W[(size_t)ks * Nout + c]       : 0.0f;
    v.y = (c < Nout) ? W[(size_t)(ks + 1) * Nout + c] : 0.0f;
    ((v2f*)Wp)[idx] = v;
}

// ---------------------------------------------------------------------------
// H[n, Nout] = A[n, K] @ W[K, Nout]  via V_WMMA_F32_16X16X4_F32 (exact fp32).
// One block = 16 output rows staged in LDS; 8 wave32, each wave owns column
// tiles ct = wave, wave+8, ...  Inner loop: ds_load_b64 + global_load_b64 +
// v_wmma, all unconditional.
// ---------------------------------------------------------------------------
__global__ void gcn_gemm_wmma_f32(const float* __restrict__ A,
                                  const float* __restrict__ Wp,
                                  float* __restrict__ H,
                                  int K, int Nout) {
    __shared__ float As[16 * KMAX];
    const int row0 = blockIdx.x * 16;
    const int tid  = threadIdx.x;

    // Cooperative float4 copy of the contiguous 16xK stripe into LDS.
    {
        const float4* gsrc = (const float4*)(A + (size_t)row0 * K);
        float4* ldst = (float4*)As;
        const int nvec = (16 * K) >> 2;
        for (int t = tid; t < nvec; t += blockDim.x) ldst[t] = gsrc[t];
    }
    __syncthreads();

    const int lane = tid & 31;
    const int wave = tid >> 5;          // 0..7
    const int half = lane >> 4;         // 0 or 1
    const int l16  = lane & 15;
    const int nColTiles = (Nout + 15) / 16;
    const int K4 = K >> 2;

    const float* arow = As + l16 * K + 2 * half;   // lane's (M,K) base in LDS
    const v2f* bp = (const v2f*)Wp;

    for (int ct = wave; ct < nColTiles; ct += 8) {
        v8f acc = {};
        const v2f* bcol = bp + (((size_t)ct * K4 * 2 + half) * 16 + l16);
        for (int k4 = 0; k4 < K4; ++k4) {
            const v2f a = *(const v2f*)(arow + (k4 << 2));
            const v2f b = bcol[(size_t)k4 * 32];
            acc = __builtin_amdgcn_wmma_f32_16x16x4_f32(
                false, a, false, b, (short)0, acc, false, false);
        }
        const int c = ct * 16 + l16;
        if (c < Nout) {
            #pragma unroll
            for (int r = 0; r < 8; ++r)
                H[(size_t)(row0 + r + 8 * half) * Nout + c] = acc[r];
        }
    }
}

// ---------------------------------------------------------------------------
// Degree / normalization helpers
// ---------------------------------------------------------------------------
__global__ void gcn_fill_f32(float* __restrict__ p, float v, int n) {
    int i = blockIdx.x * blockDim.x + threadIdx.x;
    if (i < n) p[i] = v;
}

__global__ void gcn_deg_scatter(const int* __restrict__ dst,
                                float* __restrict__ deg, int E) {
    int e = blockIdx.x * blockDim.x + threadIdx.x;
    if (e < E) atomicAdd(&deg[dst[e]], 1.0f);
}

__global__ void gcn_rsqrt(float* __restrict__ d, int n) {
    int i = blockIdx.x * blockDim.x + threadIdx.x;
    if (i < n) {
        float v = d[i];
        d[i] = (v > 0.0f) ? rsqrtf(v) : 0.0f;
    }
}

// agg[i,:] = dinv[i]^2 * h[i,:]   (self-loop term initializes the scatter)
__global__ void gcn_init_agg(const float* __restrict__ h,
                             const float* __restrict__ dinv,
                             float* __restrict__ agg, int D) {
    const int i = blockIdx.x;
    const float w = dinv[i] * dinv[i];
    const size_t base = (size_t)i * D;
    for (int f = threadIdx.x; f < D; f += blockDim.x)
        agg[base + f] = w * h[base + f];
}

// agg[dst] += dinv[src]*dinv[dst] * h[src]; block-per-edge, coalesced gather.
__global__ void gcn_edge_scatter(const int* __restrict__ src,
                                 const int* __restrict__ dst,
                                 const float* __restrict__ dinv,
                                 const float* __restrict__ h,
                                 float* __restrict__ agg, int D) {
    const int e = blockIdx.x;
    const int s = src[e];
    const int d = dst[e];
    const float w = dinv[s] * dinv[d];
    const float* hs = h + (size_t)s * D;
    float* ad = agg + (size_t)d * D;
    for (int f = threadIdx.x; f < D; f += blockDim.x)
        atomicAdd(&ad[f], w * hs[f]);
}

// out[i,:] = relu(agg[i,:] + b)
__global__ void gcn_bias_relu(const float* __restrict__ agg,
                              const float* __restrict__ b,
                              float* __restrict__ out, int D) {
    const int i = blockIdx.x;
    const size_t base = (size_t)i * D;
    for (int f = threadIdx.x; f < D; f += blockDim.x)
        out[base + f] = fmaxf(agg[base + f] + b[f], 0.0f);
}

// Global mean pool: segment sums via atomics (batch ids in [0, N_GRAPHS))
__global__ void gcn_pool(const float* __restrict__ h,
                         const int* __restrict__ batch,
                         float* __restrict__ sums,
                         float* __restrict__ cnts, int D) {
    const int i = blockIdx.x;
    const int g = batch[i];
    const float* hi = h + (size_t)i * D;
    float* sg = sums + (size_t)g * D;
    for (int f = threadIdx.x; f < D; f += blockDim.x)
        atomicAdd(&sg[f], hi[f]);
    if (threadIdx.x == 0) atomicAdd(&cnts[g], 1.0f);
}

__global__ void gcn_readout(const float* __restrict__ sums,
                            const float* __restrict__ cnts,
                            const float* __restrict__ Wm,
                            const float* __restrict__ bm,
                            float* __restrict__ out, int G, int D) {
    const int g = blockIdx.x * blockDim.x + threadIdx.x;
    if (g >= G) return;
    const float inv = 1.0f / fmaxf(cnts[g], 1.0f);
    const float* sg = sums + (size_t)g * D;
    float acc = 0.0f;
    for (int f = 0; f < D; ++f) acc += sg[f] * Wm[f];
    out[g] = acc * inv + bm[0];
}

// ---------------------------------------------------------------------------
extern "C" void kernel_launch(void* const* d_in, const int* in_sizes, int n_in,
                              void* d_out, int out_size, void* d_ws, size_t ws_size,
                              hipStream_t stream) {
    const float* x    = (const float*)d_in[0];
    const int*   eidx = (const int*)  d_in[1];
    const int*   batch= (const int*)  d_in[2];
    const float* W1   = (const float*)d_in[3];
    const float* b1   = (const float*)d_in[4];
    const float* W2   = (const float*)d_in[5];
    const float* b2   = (const float*)d_in[6];
    const float* W3   = (const float*)d_in[7];
    const float* b3   = (const float*)d_in[8];
    const float* Wm   = (const float*)d_in[9];
    const float* bm   = (const float*)d_in[10];
    float* out = (float*)d_out;

    const int N = in_sizes[2];          // 100000 nodes
    const int E = in_sizes[1] / 2;      // 800000 edges
    const int* src = eidx;
    const int* dst = eidx + E;

    // Workspace layout (floats):
    const int nct_hid = (D_HID + 15) / 16;              // 19 col tiles
    const int nct_out = D_OUT / 16;                     // 8 col tiles
    const size_t wpackElems = (size_t)nct_hid * (KMAX / 4) * 64;  // max packed W

    float* ws    = (float*)d_ws;
    float* dinv  = ws;                                  // N
    float* Wpack = ws + (((size_t)N + 63) & ~(size_t)63);
    float* bufA  = Wpack + ((wpackElems + 63) & ~(size_t)63);
    float* bufB  = bufA + (size_t)N * D_HID;
    float* sums  = bufB + (size_t)N * D_HID;            // 512 * 128
    float* cnts  = sums + (size_t)N_GRAPHS * D_OUT;     // 512

    const int T = 256;
    // --- degree + symmetric normalization (self loop: deg starts at 1) ---
    gcn_fill_f32<<<(N + T - 1) / T, T, 0, stream>>>(dinv, 1.0f, N);
    gcn_deg_scatter<<<(E + T - 1) / T, T, 0, stream>>>(dst, dinv, E);
    gcn_rsqrt<<<(N + T - 1) / T, T, 0, stream>>>(dinv, N);

    const int rowBlocks = N / 16;       // 6250
    const int packHid = nct_hid * (D_IN / 4) * 32;   // float2 slots, 300->300
    const int packOut = nct_out * (D_HID / 4) * 32;  // float2 slots, 300->128

    // --- Layer 1: x[N,300] @ W1[300,300] ---
    gcn_pack_w<<<(packHid + T - 1) / T, T, 0, stream>>>(W1, Wpack, D_IN, D_HID, nct_hid);
    gcn_gemm_wmma_f32<<<rowBlocks, 256, 0, stream>>>(x, Wpack, bufA, D_IN, D_HID);
    gcn_init_agg<<<N, 256, 0, stream>>>(bufA, dinv, bufB, D_HID);
    gcn_edge_scatter<<<E, 256, 0, stream>>>(src, dst, dinv, bufA, bufB, D_HID);
    gcn_bias_relu<<<N, 256, 0, stream>>>(bufB, b1, bufA, D_HID);

    // --- Layer 2: h1[N,300] @ W2[300,300] ---
    gcn_pack_w<<<(packHid + T - 1) / T, T, 0, stream>>>(W2, Wpack, D_HID, D_HID, nct_hid);
    gcn_gemm_wmma_f32<<<rowBlocks, 256, 0, stream>>>(bufA, Wpack, bufB, D_HID, D_HID);
    gcn_init_agg<<<N, 256, 0, stream>>>(bufB, dinv, bufA, D_HID);
    gcn_edge_scatter<<<E, 256, 0, stream>>>(src, dst, dinv, bufB, bufA, D_HID);
    gcn_bias_relu<<<N, 256, 0, stream>>>(bufA, b2, bufB, D_HID);

    // --- Layer 3: h2[N,300] @ W3[300,128] ---
    gcn_pack_w<<<(packOut + T - 1) / T, T, 0, stream>>>(W3, Wpack, D_HID, D_OUT, nct_out);
    gcn_gemm_wmma_f32<<<rowBlocks, 256, 0, stream>>>(bufB, Wpack, bufA, D_HID, D_OUT);
    gcn_init_agg<<<N, 128, 0, stream>>>(bufA, dinv, bufB, D_OUT);
    gcn_edge_scatter<<<E, 128, 0, stream>>>(src, dst, dinv, bufA, bufB, D_OUT);
    gcn_bias_relu<<<N, 128, 0, stream>>>(bufB, b3, bufA, D_OUT);

    // --- Global mean pool + readout ---
    const int poolElems = N_GRAPHS * D_OUT + N_GRAPHS;
    gcn_fill_f32<<<(poolElems + T - 1) / T, T, 0, stream>>>(sums, 0.0f, poolElems);
    gcn_pool<<<N, 128, 0, stream>>>(bufA, batch, sums, cnts, D_OUT);
    gcn_readout<<<(N_GRAPHS + T - 1) / T, T, 0, stream>>>(sums, cnts, Wm, bm, out,
                                                          N_GRAPHS, D_OUT);
    (void)n_in; (void)out_size; (void)ws_size;
}